// DGCNN_6433861010018
// MI455X (gfx1250) — compile-verified
//
#include <hip/hip_runtime.h>
#include <hip/hip_bf16.h>
#include <math.h>

// ---------------- problem constants (from reference) ----------------
#define BGRAPHS 512
#define NPG     200
#define DEG     16
#define FEAT    128
#define NTOT    (BGRAPHS * NPG)        // 102400
#define ETOT    (NTOT * DEG)           // 1638400
#define EPG     (NPG * DEG)            // 3200
#define DLAT    97                     // 32+32+32+1
#define KTOP    30
#define C1      16
#define C2      32
#define KW2     5
#define OUTF    128
#define MPAD    208                    // 200 rounded up to 16 (13 M-tiles)
#define ZSTR    36                     // z/p row stride (16B-aligned, no 64-bank hit)
#define WSTR    33                     // W row stride (odd -> lane-half conflict free)

typedef __attribute__((ext_vector_type(2))) float v2f;
typedef __attribute__((ext_vector_type(8))) float v8f;

// K=32 GEMM: zb[MPAD x 32] = Asrc[:, a_col..a_col+32) @ wbuf[32 x 32]
// (A from LDS cat buffer, B from LDS W buffer). 26 tiles over 8 waves; tile
// predicate is wave-uniform => EXEC all-ones at each WMMA.
static __device__ __forceinline__ void gemm_k32(
    const float (*Asrc)[DLAT], int a_col,
    const float (*wbuf)[WSTR],
    float (*zb)[ZSTR], int wave, int lane)
{
    const int m  = lane & 15;
    const int hf = lane >> 4;
    for (int i = 0; i < 4; ++i) {
        const int t = wave + 8 * i;
        if (t >= 26) break;                       // wave-uniform
        const int r0 = (t % 13) * 16;
        const int n0 = (t / 13) * 16;
        v8f acc = {};
#pragma unroll
        for (int k0 = 0; k0 < 32; k0 += 4) {
            const int ka = k0 + 2 * hf;
            v2f a, b;
            a.x = Asrc[r0 + m][a_col + ka];
            a.y = Asrc[r0 + m][a_col + ka + 1];
            b.x = wbuf[ka][n0 + m];
            b.y = wbuf[ka + 1][n0 + m];
            acc = __builtin_amdgcn_wmma_f32_16x16x4_f32(
                false, a, false, b, (short)0, acc, false, false);
        }
#pragma unroll
        for (int j = 0; j < 8; ++j)
            zb[r0 + j + hf * 8][n0 + m] = acc[j];
    }
}

// pooled = z + scatter(z[src] -> dst) via LDS float atomics (ds_add_f32),
// then h = tanh((pooled + b)/(deg+1)) -> cat[:, out_col..out_col+32)
static __device__ __forceinline__ void pool_act(
    const int* __restrict__ ei, int ebse, int nb,
    const float* __restrict__ node_degs, const float* __restrict__ bias,
    float (*zb)[ZSTR], float (*pb)[ZSTR], float (*cat)[DLAT],
    int out_col, int tid)
{
    for (int idx = tid; idx < NPG * 32; idx += 256)
        pb[idx >> 5][idx & 31] = zb[idx >> 5][idx & 31];
    __syncthreads();
    for (int e = tid; e < EPG; e += 256) {
        const int s = ei[ebse + e] - nb;
        const int d = ei[ETOT + ebse + e] - nb;
#pragma unroll 8
        for (int c = 0; c < 32; ++c)
            atomicAdd(&pb[d][c], zb[s][c]);
    }
    __syncthreads();
    for (int idx = tid; idx < NPG * 32; idx += 256) {
        const int i = idx >> 5, c = idx & 31;
        cat[i][out_col + c] = tanhf((pb[i][c] + bias[c]) / node_degs[nb + i]);
    }
    __syncthreads();
}

extern "C" __global__ void __launch_bounds__(256, 1)
dgcnn_fused_kernel(const float* __restrict__ node_feat,
                   const float* __restrict__ node_degs,
                   const float* __restrict__ W1, const float* __restrict__ b1,
                   const float* __restrict__ W2, const float* __restrict__ b2,
                   const float* __restrict__ W3, const float* __restrict__ b3,
                   const float* __restrict__ W4, const float* __restrict__ b4,
                   const float* __restrict__ c1w, const float* __restrict__ c1b,
                   const float* __restrict__ c2w, const float* __restrict__ c2b,
                   const float* __restrict__ outw, const float* __restrict__ outb,
                   const int*   __restrict__ ei,
                   float* __restrict__ out)
{
    __shared__ float cat[MPAD][DLAT];   // concatenated layer outputs [*,97]
    __shared__ float zb[MPAD][ZSTR];    // z = h @ W
    __shared__ float pb[MPAD][ZSTR];    // pooled accumulator / layer-1 A slab
    __shared__ float wb[FEAT][WSTR];    // staged weight matrix (<= 128 x 32)
    __shared__ int   topidx[KTOP];
    __shared__ float y1[C1][KTOP];      // conv1 out  [16][30]
    __shared__ float y1p[C1][KTOP / 2]; // maxpool    [16][15]
    __shared__ float y2[C2][11];        // conv2 out; flat == dense input [352]

    const int g    = blockIdx.x;
    const int tid  = threadIdx.x;
    const int lane = tid & 31;
    const int wave = tid >> 5;
    const int m    = lane & 15;
    const int hf   = lane >> 4;
    const int nb   = g * NPG;           // node base
    const int ebse = g * EPG;           // edge base (edges are graph-contiguous)

    // zero cat padding rows 200..207 (read as A rows by layers 2-3)
    for (int idx = tid; idx < (MPAD - NPG) * DLAT; idx += 256)
        cat[NPG + idx / DLAT][idx % DLAT] = 0.0f;
    // stage W1 [128x32] into LDS (coalesced)
    for (int idx = tid; idx < FEAT * 32; idx += 256)
        wb[idx >> 5][idx & 31] = W1[idx];
    __syncthreads();

    // ---------------- layer 1 GEMM: z = node_feat @ W1, K=128 ----------------
    // K streamed in four 32-wide slabs staged into pb via async global->LDS
    // (ASYNCcnt path); per-wave accumulators persist across slabs.
    {
        v8f acc0 = {}, acc1 = {}, acc2 = {}, acc3 = {};
        for (int ks = 0; ks < 4; ++ks) {
            // stage slab node_feat[:, 32ks .. 32ks+32) -> pb (128-bit async)
            for (int idx = tid; idx < MPAD * 8; idx += 256) {
                const int row = idx >> 3, c4 = (idx & 7) * 4;
                if (row < NPG) {
                    const float* gp =
                        node_feat + (size_t)(nb + row) * FEAT + ks * 32 + c4;
                    const unsigned lds =
                        (unsigned)(uintptr_t)(&pb[row][c4]);  // addr[31:0] == LDS offset
                    asm volatile(
                        "global_load_async_to_lds_b128 %0, %1, off"
                        :: "v"(lds), "v"(gp) : "memory");
                } else {
                    float4 zf = make_float4(0.f, 0.f, 0.f, 0.f);
                    *(float4*)&pb[row][c4] = zf;
                }
            }
            asm volatile("s_wait_asynccnt 0x0" ::: "memory");
            __syncthreads();

#pragma unroll
            for (int i = 0; i < 4; ++i) {
                const int t = wave + 8 * i;
                if (t < 26) {                          // wave-uniform
                    const int r0 = (t % 13) * 16;
                    const int n0 = (t / 13) * 16;
                    v8f acc = (i == 0) ? acc0 : (i == 1) ? acc1
                                       : (i == 2) ? acc2 : acc3;
#pragma unroll
                    for (int k0 = 0; k0 < 32; k0 += 4) {
                        const int ka = k0 + 2 * hf;
                        v2f a, b;
                        a.x = pb[r0 + m][ka];
                        a.y = pb[r0 + m][ka + 1];
                        b.x = wb[ks * 32 + ka][n0 + m];
                        b.y = wb[ks * 32 + ka + 1][n0 + m];
                        acc = __builtin_amdgcn_wmma_f32_16x16x4_f32(
                            false, a, false, b, (short)0, acc, false, false);
                    }
                    if (i == 0) acc0 = acc; else if (i == 1) acc1 = acc;
                    else if (i == 2) acc2 = acc; else acc3 = acc;
                }
            }
            __syncthreads();   // pb re-staged next slab
        }
#pragma unroll
        for (int i = 0; i < 4; ++i) {
            const int t = wave + 8 * i;
            if (t < 26) {
                const int r0 = (t % 13) * 16;
                const int n0 = (t / 13) * 16;
                const v8f acc = (i == 0) ? acc0 : (i == 1) ? acc1
                                         : (i == 2) ? acc2 : acc3;
#pragma unroll
                for (int j = 0; j < 8; ++j)
                    zb[r0 + j + hf * 8][n0 + m] = acc[j];
            }
        }
    }
    __syncthreads();
    pool_act(ei, ebse, nb, node_degs, b1, zb, pb, cat, 0, tid);

    // ---------------- layer 2 ----------------
    for (int idx = tid; idx < 32 * 32; idx += 256)
        wb[idx >> 5][idx & 31] = W2[idx];
    __syncthreads();
    gemm_k32(cat, 0, wb, zb, wave, lane);
    __syncthreads();
    pool_act(ei, ebse, nb, node_degs, b2, zb, pb, cat, 32, tid);

    // ---------------- layer 3 ----------------
    for (int idx = tid; idx < 32 * 32; idx += 256)
        wb[idx >> 5][idx & 31] = W3[idx];
    __syncthreads();
    gemm_k32(cat, 32, wb, zb, wave, lane);
    __syncthreads();
    pool_act(ei, ebse, nb, node_degs, b3, zb, pb, cat, 64, tid);

    // ---------------- layer 4: 32 -> 1 (scalar; WMMA N=1 wastes 15/16) -------
    for (int i = tid; i < NPG; i += 256) {
        float z = 0.0f;
#pragma unroll
        for (int c = 0; c < 32; ++c) z += cat[i][64 + c] * W4[c];
        zb[i][0] = z;
        pb[i][0] = z;
    }
    __syncthreads();
    for (int e = tid; e < EPG; e += 256) {
        const int s = ei[ebse + e] - nb;
        const int d = ei[ETOT + ebse + e] - nb;
        atomicAdd(&pb[d][0], zb[s][0]);
    }
    __syncthreads();
    for (int i = tid; i < NPG; i += 256)
        cat[i][96] = tanhf((pb[i][0] + b4[0]) / node_degs[nb + i]);
    __syncthreads();

    // ------------- sortpooling: stable top-30 on channel 96 (desc) -----------
    for (int i = tid; i < NPG; i += 256) {
        const float v = cat[i][96];
        int rank = 0;
        for (int j = 0; j < NPG; ++j) {
            const float u = cat[j][96];
            rank += (u > v) || (u == v && j < i);
        }
        if (rank < KTOP) topidx[rank] = i;
    }
    __syncthreads();

    // ------------- conv1 (stride D == per-row dot) + relu --------------------
    for (int idx = tid; idx < C1 * KTOP; idx += 256) {
        const int c = idx / KTOP, k = idx % KTOP;
        const int row = topidx[k];
        float acc = c1b[c];
        for (int d = 0; d < DLAT; ++d) acc += cat[row][d] * c1w[c * DLAT + d];
        y1[c][k] = fmaxf(acc, 0.0f);
    }
    __syncthreads();
    for (int idx = tid; idx < C1 * (KTOP / 2); idx += 256) {
        const int c = idx / (KTOP / 2), t = idx % (KTOP / 2);
        y1p[c][t] = fmaxf(y1[c][2 * t], y1[c][2 * t + 1]);
    }
    __syncthreads();
    for (int idx = tid; idx < C2 * 11; idx += 256) {
        const int c2 = idx / 11, t = idx % 11;
        float acc = c2b[c2];
#pragma unroll
        for (int c1 = 0; c1 < C1; ++c1)
#pragma unroll
            for (int j = 0; j < KW2; ++j)
                acc += y1p[c1][t + j] * c2w[(c2 * C1 + c1) * KW2 + j];
        y2[c2][t] = fmaxf(acc, 0.0f);
    }
    __syncthreads();

    // ------------- dense [352]->[128]; relu(relu(x)) == relu(x) --------------
    if (tid < OUTF) {
        const float* flat = &y2[0][0];  // [32][11] contiguous == reshape(B,-1)
        float acc = outb[tid];
        for (int i = 0; i < C2 * 11; ++i) acc += flat[i] * outw[i * OUTF + tid];
        out[g * OUTF + tid] = fmaxf(acc, 0.0f);
    }
}

extern "C" void kernel_launch(void* const* d_in, const int* in_sizes, int n_in,
                              void* d_out, int out_size, void* d_ws, size_t ws_size,
                              hipStream_t stream) {
    (void)in_sizes; (void)n_in; (void)out_size; (void)d_ws; (void)ws_size;
    const float* node_feat = (const float*)d_in[0];
    const float* node_degs = (const float*)d_in[1];
    const float* W1 = (const float*)d_in[2];
    const float* b1 = (const float*)d_in[3];
    const float* W2 = (const float*)d_in[4];
    const float* b2 = (const float*)d_in[5];
    const float* W3 = (const float*)d_in[6];
    const float* b3 = (const float*)d_in[7];
    const float* W4 = (const float*)d_in[8];
    const float* b4 = (const float*)d_in[9];
    const float* c1w = (const float*)d_in[10];
    const float* c1b = (const float*)d_in[11];
    const float* c2w = (const float*)d_in[12];
    const float* c2b = (const float*)d_in[13];
    const float* outw = (const float*)d_in[14];
    const float* outb = (const float*)d_in[15];
    const int*   ei   = (const int*)d_in[16];
    float* out = (float*)d_out;

    dgcnn_fused_kernel<<<BGRAPHS, 256, 0, stream>>>(
        node_feat, node_degs, W1, b1, W2, b2, W3, b3, W4, b4,
        c1w, c1b, c2w, c2b, outw, outb, ei, out);
}